// LocalDeliberationBlock_54417235640753
// MI455X (gfx1250) — compile-verified
//
#include <hip/hip_runtime.h>
#include <hip/hip_bf16.h>
#include <math.h>

// ---------------------------------------------------------------------------
// LocalDeliberationBlock on MI455X (gfx1250).
//   B=2, S=2048, D=768, CHUNK=32, LOOKBACK=128, TOPK=16.
// All GEMM-shaped work (5 projections, 24.2 GFLOP; banded q.k scores,
// 0.8 GFLOP) runs on the fp32 WMMA pipe. Tiles staged in LDS by the Tensor
// Data Mover (tensor_load_to_lds) with double-buffered ping-pong in the GEMM.
// ---------------------------------------------------------------------------

#define BB      2
#define SS      2048
#define DD      768
#define CHUNK   32
#define LOOKBK  128
#define TOPK    16
#define KC      16          // K-chunk staged in LDS per iteration

typedef float v2f __attribute__((ext_vector_type(2)));
typedef float v8f __attribute__((ext_vector_type(8)));
typedef unsigned int u32x4 __attribute__((ext_vector_type(4)));
typedef int i32x4 __attribute__((ext_vector_type(4)));
typedef int i32x8 __attribute__((ext_vector_type(8)));

#if defined(__has_builtin)
#if __has_builtin(__builtin_amdgcn_tensor_load_to_lds) && \
    __has_builtin(__builtin_amdgcn_s_wait_tensorcnt)
#define HAVE_TDM 1
#endif
#endif

#define WMMA_F32(a, b, c) \
    __builtin_amdgcn_wmma_f32_16x16x4_f32(false, (a), false, (b), (short)0, (c), false, false)

#if defined(HAVE_TDM)
// ---------------------------------------------------------------------------
// Issue one TDM 2D tile load Global->LDS (ISA 8.3/8.4 descriptor layout).
//   tile_w : elements per row (fp32); tile_h : rows
//   stride : row stride in the global tensor (elements)
//   pad    : insert 1 DWORD after every 16 DWORDs stored to LDS (stride 17)
// Wave-level op; issue from one wave, then s_wait_tensorcnt + barrier.
// 6-arg clang-23 builtin: (u32x4 g0, i32x8 g1, i32x4 g2, i32x4 g3, i32x8, cpol)
// ---------------------------------------------------------------------------
__device__ __forceinline__ void tdm_load_2d(const float* gsrc, unsigned lds_byte,
                                            unsigned tile_w, unsigned tile_h,
                                            unsigned stride, bool pad)
{
    const unsigned long long ga = (unsigned long long)gsrc;
    u32x4 g0;
    g0[0] = 1u;                                               // count=1, user D#
    g0[1] = lds_byte;                                         // lds_addr
    g0[2] = (unsigned)ga;                                     // global_addr lo
    g0[3] = (unsigned)((ga >> 32) & 0x1FFFFFFu) | (2u << 30); // ga hi | type=2
    i32x8 g1;
    unsigned w0 = (2u << 16);                                 // data_size = 4B
    if (pad) w0 |= (1u << 20) | (3u << 22) | (0u << 25);      // 16 dw + 1 dw pad
    g1[0] = (int)w0;
    g1[1] = (int)((stride & 0xFFFFu) << 16);                  // tensor_dim0 lo16
    g1[2] = (int)(((stride >> 16) & 0xFFFFu) | ((tile_h & 0xFFFFu) << 16));
    g1[3] = (int)(((tile_h >> 16) & 0xFFFFu) | (tile_w << 16));
    g1[4] = (int)tile_h;                                      // tile_dim1
    g1[5] = (int)stride;                                      // dim0_stride lo
    g1[6] = 0;
    g1[7] = 0;
    const i32x4 gz4 = {0, 0, 0, 0};
    const i32x8 gz8 = {0, 0, 0, 0, 0, 0, 0, 0};
    __builtin_amdgcn_tensor_load_to_lds(g0, g1, gz4, gz4, gz8, 0);
}
#endif

// ---------------------------------------------------------------------------
// Kernel 1: causal running mean restarted every CHUNK tokens.
// ---------------------------------------------------------------------------
__global__ __launch_bounds__(256) void phrase_mean_kernel(
    const float* __restrict__ h, float* __restrict__ ph)
{
    const int blk = blockIdx.x;
    const float* src = h  + (size_t)blk * CHUNK * DD;
    float*       dst = ph + (size_t)blk * CHUNK * DD;
    const int d0 = threadIdx.x, d1 = threadIdx.x + 256, d2 = threadIdx.x + 512;
    float s0 = 0.f, s1 = 0.f, s2 = 0.f;
    #pragma unroll 4
    for (int i = 0; i < CHUNK; ++i) {
        const float inv = 1.0f / (float)(i + 1);
        s0 += src[i * DD + d0];
        s1 += src[i * DD + d1];
        s2 += src[i * DD + d2];
        dst[i * DD + d0] = s0 * inv;
        dst[i * DD + d1] = s1 * inv;
        dst[i * DD + d2] = s2 * inv;
    }
}

// ---------------------------------------------------------------------------
// Kernel 2: C[M,N] = A[M,K]@W[K,N] + bias, fp32 WMMA, double-buffered TDM.
// 256 threads = 8 waves; block tile 64x64; wave (wm,wn) owns a 16x32 strip.
//   As: 64x16 pad-17 rows (conflict-free A reads); Bs: 16x64.
// Ping-pong: issue chunk i+1 DMA before computing chunk i; wait+barrier after.
// ---------------------------------------------------------------------------
__global__ __launch_bounds__(256) void gemm_wmma_f32(
    const float* __restrict__ A, const float* __restrict__ W,
    const float* __restrict__ bias, float* __restrict__ C,
    int M, int N, int K)
{
    const int tid  = threadIdx.x;
    const int wave = tid >> 5, lane = tid & 31;
    const int half = lane >> 4, lrow = lane & 15;
    const int wm = wave >> 1, wn = wave & 1;

    const int tiles_n = N / 64;
    const int bm = blockIdx.x / tiles_n;
    const int bn = blockIdx.x - bm * tiles_n;
    const int m0 = bm * 64, n0 = bn * 64;

    v8f acc0 = {}, acc1 = {};

#if defined(HAVE_TDM)
    __shared__ float As[2][64][17];
    __shared__ float Bs[2][KC][64];

    const float* aTile = A + (size_t)m0 * K;   // row stride K, chunk offset k0
    const float* bTile = W + n0;               // row stride N, chunk offset k0*N

    if (wave == 0) {
        tdm_load_2d(aTile, (unsigned)(unsigned long long)(void*)&As[0][0][0],
                    KC, 64, (unsigned)K, /*pad=*/true);
        tdm_load_2d(bTile, (unsigned)(unsigned long long)(void*)&Bs[0][0][0],
                    64, KC, (unsigned)N, /*pad=*/false);
    }
    __builtin_amdgcn_s_wait_tensorcnt(0);
    __syncthreads();

    int cur = 0;
    for (int k0 = 0; k0 < K; k0 += KC) {
        if (wave == 0 && k0 + KC < K) {        // prefetch next chunk via TDM
            tdm_load_2d(aTile + (k0 + KC),
                        (unsigned)(unsigned long long)(void*)&As[1 - cur][0][0],
                        KC, 64, (unsigned)K, true);
            tdm_load_2d(bTile + (size_t)(k0 + KC) * N,
                        (unsigned)(unsigned long long)(void*)&Bs[1 - cur][0][0],
                        64, KC, (unsigned)N, false);
        }
        #pragma unroll
        for (int kk = 0; kk < KC; kk += 4) {
            v2f a, b0, b1;
            const int ka = kk + 2 * half;
            a.x  = As[cur][wm * 16 + lrow][ka + 0];
            a.y  = As[cur][wm * 16 + lrow][ka + 1];
            b0.x = Bs[cur][ka + 0][wn * 32 + lrow];
            b0.y = Bs[cur][ka + 1][wn * 32 + lrow];
            b1.x = Bs[cur][ka + 0][wn * 32 + 16 + lrow];
            b1.y = Bs[cur][ka + 1][wn * 32 + 16 + lrow];
            acc0 = WMMA_F32(a, b0, acc0);
            acc1 = WMMA_F32(a, b1, acc1);
        }
        __builtin_amdgcn_s_wait_tensorcnt(0);  // next chunk landed (wave 0)
        __syncthreads();
        cur ^= 1;
    }
#else
    __shared__ float As[64][17];
    __shared__ float Bs[KC][64];
    const int ar = tid >> 2, ac = (tid & 3) * 4;
    const int br = tid >> 4, bc = (tid & 15) * 4;
    const float* __restrict__ aPtr = A + (size_t)(m0 + ar) * K + ac;
    const float* __restrict__ bPtr = W + (size_t)br * N + n0 + bc;
    for (int k0 = 0; k0 < K; k0 += KC) {
        const float4 av = *(const float4*)&aPtr[k0];
        As[ar][ac + 0] = av.x; As[ar][ac + 1] = av.y;
        As[ar][ac + 2] = av.z; As[ar][ac + 3] = av.w;
        *(float4*)&Bs[br][bc] = *(const float4*)&bPtr[(size_t)k0 * N];
        if (k0 + KC < K) {
            __builtin_prefetch(&aPtr[k0 + KC], 0, 3);
            __builtin_prefetch(&bPtr[(size_t)(k0 + KC) * N], 0, 3);
        }
        __syncthreads();
        #pragma unroll
        for (int kk = 0; kk < KC; kk += 4) {
            v2f a, b0, b1;
            const int ka = kk + 2 * half;
            a.x  = As[wm * 16 + lrow][ka + 0];
            a.y  = As[wm * 16 + lrow][ka + 1];
            b0.x = Bs[ka + 0][wn * 32 + lrow];
            b0.y = Bs[ka + 1][wn * 32 + lrow];
            b1.x = Bs[ka + 0][wn * 32 + 16 + lrow];
            b1.y = Bs[ka + 1][wn * 32 + 16 + lrow];
            acc0 = WMMA_F32(a, b0, acc0);
            acc1 = WMMA_F32(a, b1, acc1);
        }
        __syncthreads();
    }
#endif

    const int crow = m0 + wm * 16 + 8 * half;
    const int c0   = n0 + wn * 32 + lrow;
    const float bn0 = bias[c0], bn1 = bias[c0 + 16];
    #pragma unroll
    for (int r = 0; r < 8; ++r) {
        C[(size_t)(crow + r) * N + c0]      = acc0[r] + bn0;
        C[(size_t)(crow + r) * N + c0 + 16] = acc1[r] + bn1;
    }
}

// ---------------------------------------------------------------------------
// Kernel 3: banded attention for 16 consecutive tokens per block.
// 288 threads = 9 waves. Scores = K_rows[144 x 768] x q^T[768 x 16] on the
// WMMA pipe (wave w = pos-tile w). Ks staged by TDM when the window needs no
// clamping (t0 >= 128; block-uniform), else cooperative float4 loads.
// Band for token r is scores[r+l][r]; per-token top-16 uses a 128-bit taken
// mask with unrolling disabled (keeps the selection ~100 instrs, it is
// negligible work). Then 18-way softmax + weighted value mix.
// ---------------------------------------------------------------------------
__global__ __launch_bounds__(288) void attn_kernel(
    const float* __restrict__ q,  const float* __restrict__ k,
    const float* __restrict__ v,  const float* __restrict__ pk,
    const float* __restrict__ pv, float* __restrict__ out)
{
    const int b  = blockIdx.y;
    const int t0 = blockIdx.x * 16;
    const int tid  = threadIdx.x;
    const int wave = tid >> 5, lane = tid & 31;
    const int half = lane >> 4, lrow = lane & 15;
    const float scale = 0.03608439182435161f;      // 1/sqrt(768)

    __shared__ float Ks[144][17];
    __shared__ float qT[KC][16];
    __shared__ float scores[144][17];              // 18r+17l -> conflict-free
    __shared__ float phrs[16];
    __shared__ float wgt[16][TOPK + 2];
    __shared__ int   tpos[16][TOPK];

    // --- phrase scores q[t].pk[t-1]: waves 0..7, two tokens each ------------
    if (wave < 8) {
        #pragma unroll
        for (int rr = 0; rr < 2; ++rr) {
            const int r = wave * 2 + rr;
            const int t = t0 + r;
            float part = 0.f;
            if (t > 0) {
                const float* __restrict__ qrow  = q  + ((size_t)b * SS + t) * DD;
                const float* __restrict__ pkrow = pk + ((size_t)b * SS + t - 1) * DD;
                #pragma unroll 4
                for (int d = lane; d < DD; d += 32) part += qrow[d] * pkrow[d];
            }
            #pragma unroll
            for (int off = 16; off; off >>= 1) part += __shfl_xor(part, off, 32);
            if (lane == 0) phrs[r] = part * scale;
        }
    }

    // --- WMMA band scores ----------------------------------------------------
    const int  pos0   = t0 - LOOKBK;
    const bool can_tdm = (pos0 >= 0);              // block-uniform
    v8f acc = {};
    for (int k0 = 0; k0 < DD; k0 += KC) {
        __syncthreads();
#if defined(HAVE_TDM)
        if (can_tdm) {
            if (wave == 0)
                tdm_load_2d(&k[((size_t)b * SS + pos0) * DD + k0],
                            (unsigned)(unsigned long long)(void*)&Ks[0][0],
                            KC, 144, (unsigned)DD, /*pad=*/true);
        } else
#endif
        {
            #pragma unroll
            for (int i = 0; i < 2; ++i) {
                const int idx = tid + i * 288;
                const int row = idx >> 2, c4 = (idx & 3) * 4;
                int pos = pos0 + row;
                if (pos < 0) pos = 0;              // garbage row, masked later
                *(float4*)&Ks[row][c4] =
                    *(const float4*)&k[((size_t)b * SS + pos) * DD + k0 + c4];
            }
        }
        // stage q^T fragment: qT[kk][token]
        if (tid < 256) {
            const int kk = tid >> 4, r = tid & 15;
            qT[kk][r] = q[((size_t)b * SS + t0 + r) * DD + k0 + kk];
        }
#if defined(HAVE_TDM)
        __builtin_amdgcn_s_wait_tensorcnt(0);
#endif
        __syncthreads();
        #pragma unroll
        for (int kk = 0; kk < KC; kk += 4) {
            v2f a, bfr;
            const int ka = kk + 2 * half;
            a.x   = Ks[wave * 16 + lrow][ka + 0];
            a.y   = Ks[wave * 16 + lrow][ka + 1];
            bfr.x = qT[ka + 0][lrow];
            bfr.y = qT[ka + 1][lrow];
            acc = WMMA_F32(a, bfr, acc);
        }
    }
    __syncthreads();
    #pragma unroll
    for (int r = 0; r < 8; ++r)
        scores[wave * 16 + r + 8 * half][lrow] = acc[r] * scale;
    __syncthreads();

    // --- per-token top-16 + softmax (threads 0..15, one token each) ---------
    if (tid < 16) {
        const int r = tid, t = t0 + r;
        if (t > 0) {
            const float seq = scores[r + LOOKBK - 1][r];   // pos t-1
            const float phr = phrs[r];
            const int lmin = (t < LOOKBK) ? (LOOKBK - t) : 0;  // valid: l >= lmin
            // 128-bit taken mask; invalid slots pre-marked taken
            unsigned long long tk_lo = (lmin >= 64) ? ~0ull
                                     : ((lmin == 0) ? 0ull : ((1ull << lmin) - 1ull));
            unsigned long long tk_hi = (lmin > 64) ? ((1ull << (lmin - 64)) - 1ull) : 0ull;
            float tv[TOPK];
            #pragma unroll 1
            for (int m = 0; m < TOPK; ++m) {
                float best = -__builtin_inff(); int bi = -1;
                #pragma unroll 1
                for (int l = lmin; l < LOOKBK; ++l) {
                    const unsigned long long bit =
                        (l < 64) ? ((tk_lo >> l) & 1ull) : ((tk_hi >> (l - 64)) & 1ull);
                    if (bit) continue;
                    const float x = scores[r + l][r];
                    if (x > best) { best = x; bi = l; }
                }
                if (bi >= 0) {
                    if (bi < 64) tk_lo |= 1ull << bi; else tk_hi |= 1ull << (bi - 64);
                    tv[m] = best;
                } else tv[m] = -__builtin_inff();
                const int p = (bi >= 0) ? (t - LOOKBK + bi) : 0;
                tpos[r][m] = p > 0 ? p : 0;        // -inf entries get weight 0
            }
            float mx = fmaxf(seq, phr);
            #pragma unroll
            for (int m = 0; m < TOPK; ++m) mx = fmaxf(mx, tv[m]);
            float es = __expf(seq - mx), ep = __expf(phr - mx), tot = es + ep;
            float ev[TOPK];
            #pragma unroll
            for (int m = 0; m < TOPK; ++m) { ev[m] = __expf(tv[m] - mx); tot += ev[m]; }
            const float inv = 1.0f / tot;
            wgt[r][0] = es * inv;
            #pragma unroll
            for (int m = 0; m < TOPK; ++m) wgt[r][1 + m] = ev[m] * inv;
            wgt[r][TOPK + 1] = ep * inv;
        }
    }
    __syncthreads();

    // --- weighted value mix --------------------------------------------------
    for (int idx = tid; idx < 16 * DD; idx += 288) {
        const int r = idx / DD, d = idx - r * DD;
        const int t = t0 + r;
        const size_t orow = ((size_t)b * SS + t) * DD;
        if (t == 0) { out[orow + d] = 0.f; continue; }
        const size_t prev = ((size_t)b * SS + t - 1) * DD;
        float accv = wgt[r][0] * v[prev + d] + wgt[r][TOPK + 1] * pv[prev + d];
        #pragma unroll
        for (int m = 0; m < TOPK; ++m)
            accv += wgt[r][1 + m] * v[((size_t)b * SS + tpos[r][m]) * DD + d];
        out[orow + d] = accv;
    }
}

// ---------------------------------------------------------------------------
// Launcher. Workspace: 6 fp32 [B*S*D] buffers = 75.5 MB.
// Inputs: h, Wq,bq, Wk,bk, Wv,bv, Wpk,bpk, Wpv,bpv.
// ---------------------------------------------------------------------------
extern "C" void kernel_launch(void* const* d_in, const int* in_sizes, int n_in,
                              void* d_out, int out_size, void* d_ws, size_t ws_size,
                              hipStream_t stream)
{
    const float* h   = (const float*)d_in[0];
    const float* Wq  = (const float*)d_in[1];
    const float* bq  = (const float*)d_in[2];
    const float* Wk  = (const float*)d_in[3];
    const float* bk  = (const float*)d_in[4];
    const float* Wv  = (const float*)d_in[5];
    const float* bv  = (const float*)d_in[6];
    const float* Wpk = (const float*)d_in[7];
    const float* bpk = (const float*)d_in[8];
    const float* Wpv = (const float*)d_in[9];
    const float* bpv = (const float*)d_in[10];
    float* out = (float*)d_out;

    const size_t NE = (size_t)BB * SS * DD;
    float* ws = (float*)d_ws;
    float* ph = ws;
    float* q  = ws + 1 * NE;
    float* k  = ws + 2 * NE;
    float* v  = ws + 3 * NE;
    float* pk = ws + 4 * NE;
    float* pv = ws + 5 * NE;

    phrase_mean_kernel<<<BB * (SS / CHUNK), 256, 0, stream>>>(h, ph);

    const int M = BB * SS, N = DD, K = DD;
    const int blocks = (M / 64) * (N / 64);        // 64 * 12 = 768
    gemm_wmma_f32<<<blocks, 256, 0, stream>>>(h,  Wq,  bq,  q,  M, N, K);
    gemm_wmma_f32<<<blocks, 256, 0, stream>>>(h,  Wk,  bk,  k,  M, N, K);
    gemm_wmma_f32<<<blocks, 256, 0, stream>>>(h,  Wv,  bv,  v,  M, N, K);
    gemm_wmma_f32<<<blocks, 256, 0, stream>>>(ph, Wpk, bpk, pk, M, N, K);
    gemm_wmma_f32<<<blocks, 256, 0, stream>>>(ph, Wpv, bpv, pv, M, N, K);

    attn_kernel<<<dim3(SS / 16, BB), 288, 0, stream>>>(q, k, v, pk, pv, out);
}